// GINPoolNetwork_28424093565724
// MI455X (gfx1250) — compile-verified
//
#include <hip/hip_runtime.h>

// GIN pool network for MI455X (gfx1250, wave32).
// fp32 GEMMs via V_WMMA_F32_16X16X4_F32 with LDS-staged weights (ds_load_b64
// B-fragments co-issue with the VMEM A-stream); scatter/pool via
// non-returning global_atomic_add_f32.

typedef __attribute__((ext_vector_type(2))) float v2f;
typedef __attribute__((ext_vector_type(8))) float v8f;

#define N_NODES   100000
#define N_EDGES   800000
#define N_GRAPHS  128
#define UNITS     128
#define NUM_GINS  3
#define NUM_CLASSES 16
#define EPS_GIN   0.5f
#define BN_EPS    1e-3f

// ---------------------------------------------------------------- zero fill
__global__ void zero_kernel(float* __restrict__ p, long n4) {
  long i = (long)blockIdx.x * blockDim.x + threadIdx.x;
  long stride = (long)gridDim.x * blockDim.x;
  float4 z = make_float4(0.f, 0.f, 0.f, 0.f);
  for (; i < n4; i += stride) ((float4*)p)[i] = z;
}

// ------------------------------------------------- edge gather + scatter-add
// One 32-lane wave per edge; lane handles 4 consecutive columns (float4).
// agg[dst, :] += edge_weight[e] * h[src, :]
__global__ void scatter_kernel(const float* __restrict__ h, int hstride,
                               const int* __restrict__ ei,
                               const float* __restrict__ ew,
                               float* __restrict__ agg) {
  long t = (long)blockIdx.x * blockDim.x + threadIdx.x;
  int e = (int)(t >> 5);
  if (e >= N_EDGES) return;
  int lane = (int)(t & 31);
  int dst = ei[e];
  int src = ei[N_EDGES + e];
  float w = ew[e];
  const float4 v = ((const float4*)(h + (long)src * hstride))[lane];
  float* ar = agg + (long)dst * UNITS + lane * 4;
  atomicAdd(ar + 0, v.x * w);   // non-returning global_atomic_add_f32
  atomicAdd(ar + 1, v.y * w);
  atomicAdd(ar + 2, v.z * w);
  atomicAdd(ar + 3, v.w * w);
}

// Cooperative stage of a 128x128 fp32 weight matrix into LDS in
// fragment-pair layout: ldsW[p*128+col] = { W[2p][col], W[2p+1][col] }.
// 64 pairs x 128 cols x 8B = 64 KB; B fragment = one aligned ds_load_b64.
__device__ __forceinline__ void stage_weights(const float* __restrict__ W,
                                              float2* __restrict__ ldsW) {
  for (int i = threadIdx.x; i < 64 * UNITS; i += 256) {
    int p   = i >> 7;          // pair index: rows 2p, 2p+1
    int col = i & 127;
    float2 v;
    v.x = W[(long)(2 * p)     * UNITS + col];
    v.y = W[(long)(2 * p + 1) * UNITS + col];
    ldsW[i] = v;
  }
  __syncthreads();
}

// ---------------------------------------------------------------- GIN FC1
// out = relu( ((1+eps)*h + agg) @ W + b ),  M=100000, N=K=128.
// One wave per 16-row strip; 8 WMMA 16x16 tiles cover all 128 columns.
__global__ void gin_fc1_kernel(const float* __restrict__ h, int hstride,
                               const float* __restrict__ agg,
                               const float* __restrict__ W,
                               const float* __restrict__ bias,
                               float* __restrict__ out) {
  __shared__ float2 ldsW[64 * UNITS];
  stage_weights(W, ldsW);                     // all 256 threads, then barrier

  int wave = threadIdx.x >> 5;
  int lane = threadIdx.x & 31;
  int strip = blockIdx.x * 8 + wave;
  if (strip * 16 >= N_NODES) return;          // wave-uniform: EXEC stays all-1s
  int m = lane & 15;
  int half = lane >> 4;                       // lanes 0-15: K+{0,1}; 16-31: K+{2,3}
  long arow = (long)(strip * 16 + m);

  v8f c[8] = {};
  for (int k = 0; k < 32; ++k) {
    int kb = k * 4 + half * 2;
    int p  = k * 2 + half;                    // kb/2
    v2f a;
    a[0] = 1.5f * h[arow * hstride + kb    ] + agg[arow * UNITS + kb    ];
    a[1] = 1.5f * h[arow * hstride + kb + 1] + agg[arow * UNITS + kb + 1];
#pragma unroll
    for (int n = 0; n < 8; ++n) {
      int col = n * 16 + m;
      float2 w2 = ldsW[p * UNITS + col];      // ds_load_b64
      v2f b; b[0] = w2.x; b[1] = w2.y;
      c[n] = __builtin_amdgcn_wmma_f32_16x16x4_f32(
          false, a, false, b, (short)0, c[n], false, false);
    }
  }
  int mbase = strip * 16 + half * 8;          // C/D: VGPR r -> M = mbase + r
#pragma unroll
  for (int n = 0; n < 8; ++n) {
    int col = n * 16 + m;
    float bc = bias[col];
#pragma unroll
    for (int r = 0; r < 8; ++r) {
      float v = c[n][r] + bc;
      out[(long)(mbase + r) * UNITS + col] = v > 0.f ? v : 0.f;
    }
  }
}

// ---------------------------------------------------------------- GIN FC2
// hcat[:, off:off+128] = relu( BN( t @ W + b ) ), strided store (row stride 384)
__global__ void gin_fc2_kernel(const float* __restrict__ t,
                               const float* __restrict__ W,
                               const float* __restrict__ bias,
                               const float* __restrict__ gamma,
                               const float* __restrict__ beta,
                               const float* __restrict__ mean,
                               const float* __restrict__ var,
                               float* __restrict__ hcat_slice) {
  __shared__ float2 ldsW[64 * UNITS];
  stage_weights(W, ldsW);

  int wave = threadIdx.x >> 5;
  int lane = threadIdx.x & 31;
  int strip = blockIdx.x * 8 + wave;
  if (strip * 16 >= N_NODES) return;
  int m = lane & 15;
  int half = lane >> 4;
  long arow = (long)(strip * 16 + m);

  v8f c[8] = {};
  for (int k = 0; k < 32; ++k) {
    int kb = k * 4 + half * 2;
    int p  = k * 2 + half;
    v2f a;
    a[0] = t[arow * UNITS + kb];
    a[1] = t[arow * UNITS + kb + 1];
#pragma unroll
    for (int n = 0; n < 8; ++n) {
      int col = n * 16 + m;
      float2 w2 = ldsW[p * UNITS + col];
      v2f b; b[0] = w2.x; b[1] = w2.y;
      c[n] = __builtin_amdgcn_wmma_f32_16x16x4_f32(
          false, a, false, b, (short)0, c[n], false, false);
    }
  }
  int mbase = strip * 16 + half * 8;
#pragma unroll
  for (int n = 0; n < 8; ++n) {
    int col = n * 16 + m;
    float s  = gamma[col] * rsqrtf(var[col] + BN_EPS);
    float sh = beta[col] - mean[col] * s;
    float bc = bias[col];
#pragma unroll
    for (int r = 0; r < 8; ++r) {
      float v = (c[n][r] + bc) * s + sh;
      hcat_slice[(long)(mbase + r) * (NUM_GINS * UNITS) + col] = v > 0.f ? v : 0.f;
    }
  }
}

// ---------------------------------------------------------------- pooling
// pooled[g, :] += hcat[n, :]  (node_graph_index atomically scattered)
__global__ void pool_kernel(const float* __restrict__ hcat,
                            const int* __restrict__ ngi,
                            float* __restrict__ pooled) {
  long t = (long)blockIdx.x * blockDim.x + threadIdx.x;
  if (t >= (long)N_NODES * 96) return;        // 384 cols / 4 = 96 chunks
  int n  = (int)(t / 96);
  int c4 = (int)(t % 96);
  float4 v = ((const float4*)(hcat + (long)n * (NUM_GINS * UNITS)))[c4];
  int g = ngi[n];
  float* pr = pooled + (long)g * (NUM_GINS * UNITS) + c4 * 4;
  atomicAdd(pr + 0, v.x);
  atomicAdd(pr + 1, v.y);
  atomicAdd(pr + 2, v.z);
  atomicAdd(pr + 3, v.w);
}

// ---------------------------------------------------------------- MLP dense 1
// g = relu(pooled @ Wm1 + bm1), M=128 (8 waves, one block), K=384, N=128
__global__ void mlp1_kernel(const float* __restrict__ pooled,
                            const float* __restrict__ Wm1,
                            const float* __restrict__ bm1,
                            float* __restrict__ g) {
  int wave = threadIdx.x >> 5;
  int lane = threadIdx.x & 31;
  int strip = wave;                           // 8 strips * 16 rows = 128
  int m = lane & 15;
  int half = lane >> 4;
  long arow = (long)(strip * 16 + m);

  v8f c[8] = {};
  for (int k = 0; k < 96; ++k) {              // K = 384 = 96 * 4
    int kb = k * 4 + half * 2;
    v2f a;
    a[0] = pooled[arow * (NUM_GINS * UNITS) + kb];
    a[1] = pooled[arow * (NUM_GINS * UNITS) + kb + 1];
#pragma unroll
    for (int n = 0; n < 8; ++n) {
      int col = n * 16 + m;
      v2f b;
      b[0] = Wm1[(long)kb * UNITS + col];
      b[1] = Wm1[(long)(kb + 1) * UNITS + col];
      c[n] = __builtin_amdgcn_wmma_f32_16x16x4_f32(
          false, a, false, b, (short)0, c[n], false, false);
    }
  }
  int mbase = strip * 16 + half * 8;
#pragma unroll
  for (int n = 0; n < 8; ++n) {
    int col = n * 16 + m;
    float bc = bm1[col];
#pragma unroll
    for (int r = 0; r < 8; ++r) {
      float v = c[n][r] + bc;
      g[(long)(mbase + r) * UNITS + col] = v > 0.f ? v : 0.f;
    }
  }
}

// ---------------------------------------------------------------- logits
// logits = g @ Wm2 + bm2  (128 x 16, K=128 — negligible, plain VALU)
__global__ void logits_kernel(const float* __restrict__ g,
                              const float* __restrict__ Wm2,
                              const float* __restrict__ bm2,
                              float* __restrict__ out) {
  int t = blockIdx.x * blockDim.x + threadIdx.x;
  if (t >= N_GRAPHS * NUM_CLASSES) return;
  int r = t / NUM_CLASSES;
  int c = t % NUM_CLASSES;
  float acc = bm2[c];
  for (int k = 0; k < UNITS; ++k)
    acc += g[(long)r * UNITS + k] * Wm2[(long)k * NUM_CLASSES + c];
  out[t] = acc;
}

// ================================================================ launcher
extern "C" void kernel_launch(void* const* d_in, const int* in_sizes, int n_in,
                              void* d_out, int out_size, void* d_ws, size_t ws_size,
                              hipStream_t stream) {
  const float* x      = (const float*)d_in[0];
  const int*   ei     = (const int*)  d_in[1];
  const float* ew     = (const float*)d_in[2];
  const int*   ngi    = (const int*)  d_in[3];
  const float* W1     = (const float*)d_in[4];
  const float* b1     = (const float*)d_in[5];
  const float* W2     = (const float*)d_in[6];
  const float* b2     = (const float*)d_in[7];
  const float* gamma  = (const float*)d_in[8];
  const float* beta   = (const float*)d_in[9];
  const float* bnmean = (const float*)d_in[10];
  const float* bnvar  = (const float*)d_in[11];
  const float* Wm1    = (const float*)d_in[12];
  const float* bm1    = (const float*)d_in[13];
  const float* Wm2    = (const float*)d_in[14];
  const float* bm2    = (const float*)d_in[15];
  float* logits = (float*)d_out;

  // workspace carve (all fp32): agg | t1 | hcat | pooled | g  (~256 MB)
  float* agg    = (float*)d_ws;
  float* t1     = agg    + (size_t)N_NODES * UNITS;
  float* hcat   = t1     + (size_t)N_NODES * UNITS;
  float* pooled = hcat   + (size_t)N_NODES * (NUM_GINS * UNITS);
  float* gbuf   = pooled + (size_t)N_GRAPHS * (NUM_GINS * UNITS);

  const int TB = 256;
  const int strips = N_NODES / 16;                       // 6250 (exact)
  const int fc_blocks = (strips + 7) / 8;                // 782
  const int scat_blocks = (int)(((long)N_EDGES * 32 + TB - 1) / TB);   // 100000
  const int pool_blocks = (int)(((long)N_NODES * 96 + TB - 1) / TB);   // 37500

  const float* h = x;
  int hstride = UNITS;

  for (int i = 0; i < NUM_GINS; ++i) {
    zero_kernel<<<4096, TB, 0, stream>>>(agg, (long)N_NODES * UNITS / 4);
    scatter_kernel<<<scat_blocks, TB, 0, stream>>>(h, hstride, ei, ew, agg);
    gin_fc1_kernel<<<fc_blocks, TB, 0, stream>>>(
        h, hstride, agg, W1 + (size_t)i * UNITS * UNITS, b1 + (size_t)i * UNITS, t1);
    gin_fc2_kernel<<<fc_blocks, TB, 0, stream>>>(
        t1, W2 + (size_t)i * UNITS * UNITS, b2 + (size_t)i * UNITS,
        gamma + (size_t)i * UNITS, beta + (size_t)i * UNITS,
        bnmean + (size_t)i * UNITS, bnvar + (size_t)i * UNITS,
        hcat + (size_t)i * UNITS);
    h = hcat + (size_t)i * UNITS;                        // next layer reads hcat slice
    hstride = NUM_GINS * UNITS;                          // 384
  }

  zero_kernel<<<48, TB, 0, stream>>>(pooled, (long)N_GRAPHS * (NUM_GINS * UNITS) / 4);
  pool_kernel<<<pool_blocks, TB, 0, stream>>>(hcat, ngi, pooled);
  mlp1_kernel<<<1, TB, 0, stream>>>(pooled, Wm1, bm1, gbuf);
  logits_kernel<<<(N_GRAPHS * NUM_CLASSES + TB - 1) / TB, TB, 0, stream>>>(
      gbuf, Wm2, bm2, logits);
}